// LIFCell_26877905338534
// MI455X (gfx1250) — compile-verified
//
#include <hip/hip_runtime.h>
#include <stdint.h>

// LIF recurrence: v = v*sigmoid(decay_logit)*(1-z) + x_t ; z = (v > 0.5)
// Shapes: x[B=256][T=1000][H=256] f32, decay_logit[H] f32 -> spikes[B][T][H] f32.
// Memory-bound streaming scan over T; gfx1250 async-to-LDS pipeline (DEPTH ahead).

#define LIF_B 256
#define LIF_T 1000
#define LIF_H 256

#define DEPTH 32          // async issue-ahead distance (<= 63, ASYNCcnt is 6 bits)
#define SLOTS 64          // LDS circular buffer slots = 2*DEPTH (WAR safety margin)
#define TPB   128         // threads per block (4 wave32s)
#define GROUP 4           // steps consumed per wait; (LIF_T - DEPTH) % GROUP == 0

__global__ __launch_bounds__(TPB) void lif_async_kernel(
    const float* __restrict__ x,
    const float* __restrict__ decay_logit,
    float* __restrict__ out)
{
  __shared__ float stage[SLOTS * TPB];   // 64 * 128 * 4B = 32 KB

  const int tid = threadIdx.x;
  const int gid = blockIdx.x * TPB + tid;      // 0 .. B*H-1
  const int b   = gid >> 8;                    // H == 256
  const int h   = gid & 255;

  // gfx1250: flat address of LDS keeps the LDS byte offset in addr[31:0]
  // (aperture tag lives entirely in addr[63:32]); truncation yields the
  // wave-relative LDS address the async-load VDST VGPR expects.
  const unsigned lds_base = (unsigned)(uintptr_t)(&stage[tid]);

  const size_t row = (size_t)b * (LIF_T * LIF_H) + (size_t)h;  // &x[b][0][h]
  const float* gp = x + row;
  float* op = out + row;

  const float dl    = decay_logit[h];
  const float decay = 1.0f / (1.0f + expf(-dl));   // sigmoid (full precision)

  // ---- prologue: fill the async pipeline DEPTH steps deep ----
  #pragma unroll
  for (int t = 0; t < DEPTH; ++t) {
    unsigned laddr = lds_base + (unsigned)((t & (SLOTS - 1)) * TPB * 4);
    unsigned long long ga = (unsigned long long)(uintptr_t)(gp + (size_t)t * LIF_H);
    asm volatile("global_load_async_to_lds_b32 %0, %1, off"
                 :: "v"(laddr), "v"(ga)
                 : "memory");
  }

  float v = 0.0f;
  float z = 0.0f;

  // ---- steady state: one wait per GROUP of 4 steps ----
  // Before consuming t..t+3 we have issued through t+DEPTH-1; requiring
  // ASYNCcnt <= DEPTH-GROUP (28) guarantees loads t..t+3 have landed.
  for (int t = 0; t < LIF_T - DEPTH; t += GROUP) {
    asm volatile("s_wait_asynccnt 28" ::: "memory");

    #pragma unroll
    for (int j = 0; j < GROUP; ++j) {
      float xt = stage[((t + j) & (SLOTS - 1)) * TPB + tid];
      float c = (z > 0.0f) ? 0.0f : decay;   // soft reset gates the decay path
      v = __builtin_fmaf(v, c, xt);          // v = v*decay*(1-z) + x_t
      z = (v > 0.5f) ? 1.0f : 0.0f;          // heaviside(v - 0.5)
      op[(size_t)(t + j) * LIF_H] = z;
    }

    // refill: issue prefetches for steps t+DEPTH .. t+DEPTH+3
    #pragma unroll
    for (int j = 0; j < GROUP; ++j) {
      unsigned laddr = lds_base +
          (unsigned)(((t + DEPTH + j) & (SLOTS - 1)) * TPB * 4);
      unsigned long long ga =
          (unsigned long long)(uintptr_t)(gp + (size_t)(t + DEPTH + j) * LIF_H);
      asm volatile("global_load_async_to_lds_b32 %0, %1, off"
                   :: "v"(laddr), "v"(ga)
                   : "memory");
    }
  }

  // ---- drain: everything already issued; wait all, then finish last DEPTH ----
  asm volatile("s_wait_asynccnt 0" ::: "memory");
  #pragma unroll 4
  for (int t = LIF_T - DEPTH; t < LIF_T; ++t) {
    float xt = stage[(t & (SLOTS - 1)) * TPB + tid];
    float c = (z > 0.0f) ? 0.0f : decay;
    v = __builtin_fmaf(v, c, xt);
    z = (v > 0.5f) ? 1.0f : 0.0f;
    op[(size_t)t * LIF_H] = z;
  }
}

extern "C" void kernel_launch(void* const* d_in, const int* in_sizes, int n_in,
                              void* d_out, int out_size, void* d_ws, size_t ws_size,
                              hipStream_t stream) {
  (void)in_sizes; (void)n_in; (void)d_ws; (void)ws_size; (void)out_size;

  const float* x           = (const float*)d_in[0];   // [B,T,H] f32
  const float* decay_logit = (const float*)d_in[1];   // [H] f32
  float* out               = (float*)d_out;           // [B,T,H] f32

  const int total  = LIF_B * LIF_H;                   // one thread per (b,h)
  const int blocks = total / TPB;                     // 512 blocks of 128 threads

  lif_async_kernel<<<blocks, TPB, 0, stream>>>(x, decay_logit, out);
}